// LinkPredictor_40269613367575
// MI455X (gfx1250) — compile-verified
//
#include <hip/hip_runtime.h>
#include <hip/hip_bf16.h>

// ---------------------------------------------------------------------------
// RGCN link predictor for MI455X (gfx1250), wave32, WMMA f32 16x16x4.
//
// Pipeline:
//   x0 = scatter_rows(emb @ Wp + bp, node_indices)
//   for each layer L:
//     agg[n,r,:]  = mean over edges (src->n, rel=r) of x[src]      (scatter+normalize)
//     x' = relu?( concat(agg[n,0..R-1,:], x[n,:]) @ stack(W_L, root_L) + b_L )
//   scores[t] = sum_o out[s]*rel_emb[r]*out[d]   over pos++neg triples
// ---------------------------------------------------------------------------

typedef __attribute__((ext_vector_type(2))) float v2f;
typedef __attribute__((ext_vector_type(8))) float v8f;

static constexpr int N_NODES = 50000;
static constexpr int N_EDGES = 800000;
static constexpr int N_REL   = 8;
static constexpr int EMB     = 512;
static constexpr int PROJ    = 256;
static constexpr int HID     = 256;
static constexpr int OUTD    = 200;

// ---------------------------------------------------------------- utilities
__global__ void fill_zero_kernel(float* __restrict__ p, size_t n) {
  size_t stride = (size_t)gridDim.x * blockDim.x;
  for (size_t i = (size_t)blockIdx.x * blockDim.x + threadIdx.x; i < n; i += stride)
    p[i] = 0.0f;
}

// count edges per (dst, rel) segment
__global__ void count_edges_kernel(const int* __restrict__ dst,
                                   const int* __restrict__ rel,
                                   float* __restrict__ cnt, int E) {
  int stride = gridDim.x * blockDim.x;
  for (int e = blockIdx.x * blockDim.x + threadIdx.x; e < E; e += stride)
    atomicAdd(&cnt[(size_t)dst[e] * N_REL + rel[e]], 1.0f);
}

// cnt <- 1 / max(cnt, 1)
__global__ void invert_cnt_kernel(float* __restrict__ cnt, int n) {
  int stride = gridDim.x * blockDim.x;
  for (int i = blockIdx.x * blockDim.x + threadIdx.x; i < n; i += stride)
    cnt[i] = 1.0f / fmaxf(cnt[i], 1.0f);
}

// agg[(dst*R+rel)*in + c] += x[src*in + c]   (one wave32 per edge)
__global__ void scatter_agg_kernel(const float* __restrict__ x,
                                   const int* __restrict__ src,
                                   const int* __restrict__ dst,
                                   const int* __restrict__ rel,
                                   float* __restrict__ agg, int E, int in) {
  int lane = threadIdx.x & 31;
  int e = (blockIdx.x * blockDim.x + threadIdx.x) >> 5;
  if (e >= E) return;
  int s = src[e], d = dst[e], r = rel[e];
  const float* xs = x + (size_t)s * in;
  float* ag = agg + ((size_t)d * N_REL + r) * in;
  for (int c = lane; c < in; c += 32)
    atomicAdd(ag + c, xs[c]);
}

// agg[idx] *= inv_cnt[idx / in]      (in == 256 -> idx >> 8)
__global__ void normalize_agg_kernel(float* __restrict__ agg,
                                     const float* __restrict__ inv_cnt,
                                     size_t total) {
  size_t stride = (size_t)gridDim.x * blockDim.x;
  for (size_t i = (size_t)blockIdx.x * blockDim.x + threadIdx.x; i < total; i += stride)
    agg[i] *= inv_cnt[i >> 8];
}

// Pack stacked B = [W (KW rows); root (Ktot-KW rows)] (row-major [K, out]) into
// pair-interleaved float2 P[k2][outPad]: P[k2][o] = (B[2k2][o], B[2k2+1][o]).
// Columns o >= out are zero-padded so the GEMM can run on a padded grid.
__global__ void pack_b_kernel(const float* __restrict__ W, int KW,
                              const float* __restrict__ root,
                              int Ktot, int out, int outPad,
                              v2f* __restrict__ P) {
  size_t total = (size_t)(Ktot / 2) * outPad;
  size_t stride = (size_t)gridDim.x * blockDim.x;
  for (size_t idx = (size_t)blockIdx.x * blockDim.x + threadIdx.x; idx < total; idx += stride) {
    int k2 = (int)(idx / outPad);
    int o  = (int)(idx % outPad);
    v2f val = {};
    if (o < out) {
      int ka = 2 * k2, kb = ka + 1;
      val.x = (ka < KW) ? W[(size_t)ka * out + o] : root[(size_t)(ka - KW) * out + o];
      val.y = (kb < KW) ? W[(size_t)kb * out + o] : root[(size_t)(kb - KW) * out + o];
    }
    P[idx] = val;
  }
}

// ----------------------------------------------------------- fused WMMA GEMM
// Y[m, :] = concat(Aagg[m, 0:Kagg], Ax[m, 0:Kx]) @ Bpacked + bias   (+relu)
// One 128-thread block per 16-row tile; wave w owns 64 output columns
// (4 x 16x16 f32 accumulators), looping K in steps of 4 with V_WMMA_F32_16X16X4_F32.
// A fragment per lane = one b64 load (row K-contiguous); B fragment = one b64
// load from the pair-interleaved pack. outPad must equal 256 (4 waves x 64).
__global__ void __launch_bounds__(128)
rgcn_wmma_gemm_kernel(const float* __restrict__ Aagg, int Kagg,
                      const float* __restrict__ Ax, int Kx,
                      const v2f* __restrict__ Bp, int outPad,
                      const float* __restrict__ bias,
                      float* __restrict__ y, int out, int nrows,
                      const int* __restrict__ scatterIdx, int doRelu) {
  const int lane   = threadIdx.x & 31;
  const int waveId = threadIdx.x >> 5;
  const int hi     = lane >> 4;   // K-half selector within the fragment
  const int lo     = lane & 15;
  const int tmBase = blockIdx.x * 16;
  const int colBase = waveId * 64;
  const int arow   = tmBase + lo;            // A row owned by this lane

  const float* aggRow = Aagg + (size_t)arow * Kagg;
  const float* xRow   = Ax   + (size_t)arow * Kx;

  v8f acc[4] = {};

  // Phase 1: aggregated per-(relation) inputs, K-contiguous per row.
  for (int k = 0; k < Kagg; k += 4) {
    v2f a = *(const v2f*)(aggRow + k + 2 * hi);
    const v2f* brow = Bp + (size_t)((k >> 1) + hi) * outPad + colBase + lo;
#pragma unroll
    for (int t = 0; t < 4; ++t) {
      v2f b = brow[t * 16];
      acc[t] = __builtin_amdgcn_wmma_f32_16x16x4_f32(
          false, a, false, b, (short)0, acc[t], false, false);
    }
  }
  // Phase 2: root-transform K segment reads the layer input x directly.
  for (int k = 0; k < Kx; k += 4) {
    v2f a = *(const v2f*)(xRow + k + 2 * hi);
    int kk = Kagg + k;
    const v2f* brow = Bp + (size_t)((kk >> 1) + hi) * outPad + colBase + lo;
#pragma unroll
    for (int t = 0; t < 4; ++t) {
      v2f b = brow[t * 16];
      acc[t] = __builtin_amdgcn_wmma_f32_16x16x4_f32(
          false, a, false, b, (short)0, acc[t], false, false);
    }
  }

  // Epilogue: C layout — VGPR v holds row (v + 8*hi), col = lo.
#pragma unroll
  for (int v = 0; v < 8; ++v) {
    int m = tmBase + v + 8 * hi;
    if (m >= nrows) continue;
    int orow = scatterIdx ? scatterIdx[m] : m;
#pragma unroll
    for (int t = 0; t < 4; ++t) {
      int col = colBase + t * 16 + lo;
      if (col < out) {
        float val = acc[t][v] + bias[col];
        if (doRelu) val = fmaxf(val, 0.0f);
        y[(size_t)orow * out + col] = val;
      }
    }
  }
}

// --------------------------------------------------------- DistMult decoder
// One wave32 per triple; lanes stride the 200-dim feature, __shfl_xor reduce.
__global__ void distmult_kernel(const float* __restrict__ nodeOut,
                                const float* __restrict__ rel_emb,
                                const int* __restrict__ src,
                                const int* __restrict__ dst,
                                const int* __restrict__ rel,
                                const int* __restrict__ nsrc,
                                const int* __restrict__ ndst,
                                const int* __restrict__ nrel,
                                float* __restrict__ scores, int E, int outDim) {
  int lane = threadIdx.x & 31;
  int t = (blockIdx.x * blockDim.x + threadIdx.x) >> 5;
  if (t >= 2 * E) return;
  int s, d, r;
  if (t < E) { s = src[t]; d = dst[t]; r = rel[t]; }
  else       { int u = t - E; s = nsrc[u]; d = ndst[u]; r = nrel[u]; }
  const float* ps = nodeOut + (size_t)s * outDim;
  const float* pd = nodeOut + (size_t)d * outDim;
  const float* pr = rel_emb + (size_t)r * outDim;
  float sum = 0.0f;
  for (int o = lane; o < outDim; o += 32)
    sum += ps[o] * pr[o] * pd[o];
#pragma unroll
  for (int m = 16; m >= 1; m >>= 1)
    sum += __shfl_xor(sum, m, 32);
  if (lane == 0) scores[t] = sum;
}

// ---------------------------------------------------------------------------
extern "C" void kernel_launch(void* const* d_in, const int* in_sizes, int n_in,
                              void* d_out, int out_size, void* d_ws, size_t ws_size,
                              hipStream_t stream) {
  (void)in_sizes; (void)n_in; (void)out_size; (void)ws_size;

  const float* emb     = (const float*)d_in[0];
  const int*   nidx    = (const int*)  d_in[1];
  const int*   src     = (const int*)  d_in[2];
  const int*   dst     = (const int*)  d_in[3];
  const int*   rel     = (const int*)  d_in[4];
  const int*   nsrc    = (const int*)  d_in[5];
  const int*   ndst    = (const int*)  d_in[6];
  const int*   nrel    = (const int*)  d_in[7];
  const float* Wp      = (const float*)d_in[8];
  const float* bp      = (const float*)d_in[9];
  const float* W0      = (const float*)d_in[10];
  const float* root0   = (const float*)d_in[11];
  const float* b0      = (const float*)d_in[12];
  const float* W1      = (const float*)d_in[13];
  const float* root1   = (const float*)d_in[14];
  const float* b1      = (const float*)d_in[15];
  const float* W2      = (const float*)d_in[16];
  const float* root2   = (const float*)d_in[17];
  const float* b2      = (const float*)d_in[18];
  const float* rel_emb = (const float*)d_in[19];

  float* out_nodes = (float*)d_out;                          // [N, OUT]
  float* scores    = out_nodes + (size_t)N_NODES * OUTD;     // [2E]

  // ---- workspace carve-up (bytes, 256-aligned) ----
  char* ws = (char*)d_ws;
  size_t off = 0;
  auto carve = [&](size_t bytes) -> char* {
    char* p = ws + off;
    off += (bytes + 255) & ~(size_t)255;
    return p;
  };
  float* x0  = (float*)carve((size_t)N_NODES * 256 * 4);          //  51.2 MB
  float* x1  = (float*)carve((size_t)N_NODES * 256 * 4);          //  51.2 MB
  float* agg = (float*)carve((size_t)N_NODES * N_REL * 256 * 4);  // 409.6 MB
  float* cnt = (float*)carve((size_t)N_NODES * N_REL * 4);        //   1.6 MB
  v2f*   Bp  = (v2f*)  carve((size_t)(2304 / 2) * 256 * 8);       //   2.4 MB

  const size_t aggElems = (size_t)N_NODES * N_REL * 256;
  const int segs = N_NODES * N_REL;
  const int rowTiles = (N_NODES + 15) / 16;   // 3125

  // ---- edge-count structure (fixed across layers): cnt = 1/max(deg,1) ----
  fill_zero_kernel<<<2048, 256, 0, stream>>>(cnt, (size_t)segs);
  count_edges_kernel<<<(N_EDGES + 255) / 256, 256, 0, stream>>>(dst, rel, cnt, N_EDGES);
  invert_cnt_kernel<<<(segs + 255) / 256, 256, 0, stream>>>(cnt, segs);

  // ---- projection: x0[nidx[i], :] = emb[i] @ Wp + bp ----
  fill_zero_kernel<<<4096, 256, 0, stream>>>(x0, (size_t)N_NODES * PROJ);
  {
    int Ktot = EMB;  // 512, no agg phase
    pack_b_kernel<<<((Ktot / 2) * 256 + 255) / 256, 256, 0, stream>>>(
        Wp, Ktot, Wp, Ktot, PROJ, 256, Bp);
    rgcn_wmma_gemm_kernel<<<rowTiles, 128, 0, stream>>>(
        emb, 0, emb, EMB, Bp, 256, bp, x0, PROJ, N_NODES, nidx, /*relu=*/0);
  }

  // ---- RGCN layers ----
  struct Layer {
    const float* W; const float* root; const float* bias;
    const float* xin; float* xout; int out; int relu;
  };
  Layer layers[3] = {
      {W0, root0, b0, x0, x1,        HID,  1},
      {W1, root1, b1, x1, x0,        HID,  1},
      {W2, root2, b2, x0, out_nodes, OUTD, 0},
  };

  for (int L = 0; L < 3; ++L) {
    const Layer& ly = layers[L];
    const int in = 256;                  // PROJ == HID == 256
    const int Kagg = N_REL * in;         // 2048
    const int Ktot = Kagg + in;          // 2304

    fill_zero_kernel<<<8192, 256, 0, stream>>>(agg, aggElems);
    scatter_agg_kernel<<<(N_EDGES * 32 + 255) / 256, 256, 0, stream>>>(
        ly.xin, src, dst, rel, agg, N_EDGES, in);
    normalize_agg_kernel<<<8192, 256, 0, stream>>>(agg, cnt, aggElems);
    pack_b_kernel<<<((Ktot / 2) * 256 + 255) / 256, 256, 0, stream>>>(
        ly.W, Kagg, ly.root, Ktot, ly.out, 256, Bp);
    rgcn_wmma_gemm_kernel<<<rowTiles, 128, 0, stream>>>(
        agg, Kagg, ly.xin, in, Bp, 256, ly.bias,
        ly.xout, ly.out, N_NODES, nullptr, ly.relu);
  }

  // ---- DistMult scores over [pos ++ neg] triples ----
  {
    long long waves = 2LL * N_EDGES;
    long long threads = waves * 32;
    int blocks = (int)((threads + 255) / 256);
    distmult_kernel<<<blocks, 256, 0, stream>>>(
        out_nodes, rel_emb, src, dst, rel, nsrc, ndst, nrel,
        scores, N_EDGES, OUTD);
  }
}